// TRFAligner_47382079209934
// MI455X (gfx1250) — compile-verified
//
#include <hip/hip_runtime.h>
#include <stdint.h>

// ---------------------------------------------------------------------------
// TRF overlap-add (fold / col2im) for MI455X (gfx1250).
// out[d, t] = sum over (seq,k) with sourceIdx[seq]+k == t of TRFs[seq,k,d]
// Bandwidth-bound (~320 MB @ 23.3 TB/s => ~14us floor). Strategy:
//   - gather formulation (no atomics): one block per 32 output times
//   - binary search sorted sourceIdx for contributing windows
//   - contiguous <=16KB chunk per window staged via ASYNC global->LDS copies,
//     double-buffered with s_wait_asynccnt (CDNA5 async path)
//   - padded LDS rows (132 floats) for near-conflict-free shifted reads
//   - per-thread register accumulators, coalesced stores along t
// ---------------------------------------------------------------------------

#define N_WIN    128
#define OUT_DIM  128
#define TILE_T   32
#define ROW_F    132                      // padded LDS row stride (floats), 528B, 16B aligned
#define BUF_F    (TILE_T * ROW_F)         // 4224 floats per buffer
#define NTHREADS 256                      // 8 waves (wave32)

typedef float f32x4 __attribute__((ext_vector_type(4)));

// Types matching the builtin prototype reported by clang:
//   __builtin_amdgcn_global_load_async_to_lds_b128(v4i __device__*, v4i __shared__*, int, int)
typedef int v4i __attribute__((vector_size(16)));
typedef __attribute__((address_space(1))) v4i* g_v4i_p;   // global (AS1)
typedef __attribute__((address_space(3))) v4i* l_v4i_p;   // LDS    (AS3)

__device__ __forceinline__ void async_copy_b128(const void* g, void* l) {
#if __has_builtin(__builtin_amdgcn_global_load_async_to_lds_b128)
  __builtin_amdgcn_global_load_async_to_lds_b128((g_v4i_p)g, (l_v4i_p)l, 0, 0);
#else
  asm volatile("global_load_async_to_lds_b128 %0, %1, off"
               :
               : "v"((uint32_t)(uintptr_t)(l_v4i_p)l),
                 "v"((uint64_t)(uintptr_t)g)
               : "memory");
#endif
}

template <int N>
__device__ __forceinline__ void wait_async() {
#if __has_builtin(__builtin_amdgcn_s_wait_asynccnt)
  __builtin_amdgcn_s_wait_asynccnt(N);
#else
  asm volatile("s_wait_asynccnt %0" : : "i"(N) : "memory");
#endif
}

__device__ __forceinline__ int lower_bound_dev(const int* __restrict__ a, int n, int key) {
  int lo = 0, hi = n;
  while (lo < hi) {
    int mid = (lo + hi) >> 1;
    if (a[mid] < key) lo = mid + 1; else hi = mid;
  }
  return lo;
}

__global__ __launch_bounds__(NTHREADS)
void trf_fold_kernel(const float* __restrict__ trf,
                     const int* __restrict__ srcIdx,
                     float* __restrict__ out,
                     int nSeq, int nOut) {
  __shared__ __align__(16) float buf[2][BUF_F];
  __shared__ __align__(16) float dummy[4];

  const int tid  = threadIdx.x;
  const int lane = tid & 31;   // t within tile
  const int dgrp = tid >> 5;   // wave id -> 16-wide d group
  const int T    = blockIdx.x * TILE_T;
  const int t    = T + lane;

  // Contributing windows: sourceIdx s with  T - (N_WIN-1) <= s <= T + TILE_T - 1
  const int seqLo = lower_bound_dev(srcIdx, nSeq, T - (N_WIN - 1));
  const int seqHi = lower_bound_dev(srcIdx, nSeq, T + TILE_T);

  // stage chunk TRFs[seq, kLo .. kLo+rows-1, :] into buf[bsel] (padded rows).
  // Fixed 4 async b128 issues per thread (=> fixed ASYNCcnt of 4 per wave).
  auto stage = [&](int seq, int bsel, int kLo, int rows) {
    const char* gbase =
        (const char*)(trf + ((size_t)seq * N_WIN + kLo) * OUT_DIM);
    const int bytes = rows * (OUT_DIM * 4);
    char* lbase = (char*)&buf[bsel][0];
#pragma unroll
    for (int j = 0; j < 4; ++j) {
      const int off = (j * NTHREADS + tid) * 16;   // 16B units across chunk
      const void* g;
      void* l;
      if (off < bytes) {
        const int row    = off >> 9;               // / 512B per source row
        const int within = off & 511;
        g = gbase + off;
        l = lbase + row * (ROW_F * 4) + within;    // padded row stride 528B
      } else {                                     // keep instruction count fixed
        g = gbase;
        l = (void*)dummy;
      }
      async_copy_b128(g, l);
    }
  };

  auto window = [&](int s, int& kLo, int& rows) {
    int a = T - s;                 if (a < 0) a = 0;
    int b = T + TILE_T - 1 - s;    if (b > N_WIN - 1) b = N_WIN - 1;
    const int c = nOut - 1 - s;    if (b > c) b = c;   // discard t >= nOut
    kLo  = a;
    rows = b - a + 1;
  };

  f32x4 a0 = {0.f, 0.f, 0.f, 0.f};
  f32x4 a1 = {0.f, 0.f, 0.f, 0.f};
  f32x4 a2 = {0.f, 0.f, 0.f, 0.f};
  f32x4 a3 = {0.f, 0.f, 0.f, 0.f};

  int sCur = 0, kCur = 0, rCur = 0;
  if (seqLo < seqHi) {
    sCur = srcIdx[seqLo];
    window(sCur, kCur, rCur);
    stage(seqLo, 0, kCur, rCur);
  }

  for (int i = seqLo; i < seqHi; ++i) {
    const int bsel = (i - seqLo) & 1;
    int sN = 0, kN = 0, rN = 0;
    if (i + 1 < seqHi) {                 // prefetch next window into other buffer
      sN = srcIdx[i + 1];
      window(sN, kN, rN);
      stage(i + 1, bsel ^ 1, kN, rN);
      wait_async<4>();                   // in-order: current buffer's 4 ops done
    } else {
      wait_async<0>();
    }
    __syncthreads();                     // whole buffer visible to all waves

    // accumulate: this thread's time t maps to row (t - s - kLo) of the chunk
    const int row = t - sCur - kCur;
    if (row >= 0 && row < rCur) {
      const f32x4* p4 =
          (const f32x4*)(&buf[bsel][row * ROW_F + dgrp * 16]);
      a0 += p4[0];
      a1 += p4[1];
      a2 += p4[2];
      a3 += p4[3];
    }
    __syncthreads();                     // buffer free for overwrite (i+2 stage)

    sCur = sN; kCur = kN; rCur = rN;
  }

  // Coalesced stores: lanes are contiguous t within each d row.
  if (t < nOut) {
    const float v[16] = {a0.x, a0.y, a0.z, a0.w,
                         a1.x, a1.y, a1.z, a1.w,
                         a2.x, a2.y, a2.z, a2.w,
                         a3.x, a3.y, a3.z, a3.w};
    const int dbase = dgrp * 16;
#pragma unroll
    for (int q = 0; q < 16; ++q) {
      out[(size_t)(dbase + q) * nOut + t] = v[q];
    }
  }
}

extern "C" void kernel_launch(void* const* d_in, const int* in_sizes, int n_in,
                              void* d_out, int out_size, void* d_ws, size_t ws_size,
                              hipStream_t stream) {
  const float* trf    = (const float*)d_in[0];   // (nSeq, 128, 128) f32
  const int*   srcIdx = (const int*)d_in[1];     // (nSeq,) sorted
  float*       out    = (float*)d_out;           // (128, nRealLen) f32

  const int nSeq = in_sizes[1];
  const int nOut = out_size / OUT_DIM;           // nRealLen
  const int tiles = (nOut + TILE_T - 1) / TILE_T;

  trf_fold_kernel<<<tiles, NTHREADS, 0, stream>>>(trf, srcIdx, out, nSeq, nOut);
}